// AttentionLayer_78082505441867
// MI455X (gfx1250) — compile-verified
//
#include <hip/hip_runtime.h>
#include <math.h>

#define BB 128
#define NN 512
#define HH 256
#define NH (NN * HH)

typedef __attribute__((ext_vector_type(2))) float v2f;
typedef __attribute__((ext_vector_type(8))) float v8f;

__device__ __forceinline__ float wave_sum(float v) {
#pragma unroll
  for (int off = 16; off > 0; off >>= 1) v += __shfl_xor(v, off, 32);
  return v;
}

// K0: c1 = Wq.Wk, c2 = Wq.bk, c3 = bq.Wk, c4 = bq.bk  (one wave)
__global__ void k_consts(const float* __restrict__ Wq, const float* __restrict__ bq,
                         const float* __restrict__ Wk, const float* __restrict__ bk,
                         float* __restrict__ c) {
  int lane = threadIdx.x;
  float s1 = 0.f, s2 = 0.f, s3 = 0.f, s4 = 0.f;
  for (int h = lane; h < HH; h += 32) {
    float wq = Wq[h], b_q = bq[h], wk = Wk[h], b_k = bk[h];
    s1 += wq * wk; s2 += wq * b_k; s3 += b_q * wk; s4 += b_q * b_k;
  }
  s1 = wave_sum(s1); s2 = wave_sum(s2); s3 = wave_sum(s3); s4 = wave_sum(s4);
  if (lane == 0) { c[0] = s1; c[1] = s2; c[2] = s3; c[3] = s4; }
}

// K1: AT[n*512+j] = sum_h Wout[j, n*256+h]*Wv[h]; BT likewise with bv.
// One wave per (j,n); lane-parallel float4 loads over the contiguous 1KB row.
__global__ void k_wout(const float* __restrict__ Wout, const float* __restrict__ Wv,
                       const float* __restrict__ bv, float* __restrict__ AT,
                       float* __restrict__ BT) {
  unsigned wid = (blockIdx.x * blockDim.x + threadIdx.x) >> 5;
  int lane = threadIdx.x & 31;
  int j = wid & (NN - 1);
  int n = wid >> 9;
  const float4* p = (const float4*)(Wout + (size_t)j * NH + (size_t)n * HH);
  const float4* wv4 = (const float4*)Wv;
  const float4* bv4 = (const float4*)bv;
  float4 w0 = p[lane], w1 = p[lane + 32];
  float4 a0 = wv4[lane], a1 = wv4[lane + 32];
  float4 c0 = bv4[lane], c1 = bv4[lane + 32];
  float sA = w0.x * a0.x + w0.y * a0.y + w0.z * a0.z + w0.w * a0.w +
             w1.x * a1.x + w1.y * a1.y + w1.z * a1.z + w1.w * a1.w;
  float sB = w0.x * c0.x + w0.y * c0.y + w0.z * c0.z + w0.w * c0.w +
             w1.x * c1.x + w1.y * c1.y + w1.z * c1.z + w1.w * c1.w;
  sA = wave_sum(sA);
  sB = wave_sum(sB);
  if (lane == 0) { AT[n * NN + j] = sA; BT[n * NN + j] = sB; }
}

// K2: per-(b,m) online softmax stats over n. val = mask[b,n,m]*((c1*xn+c3)*xm + (c2*xn+c4))
__global__ void k_colstats(const float* __restrict__ X, const float* __restrict__ mask,
                           const float* __restrict__ c, float* __restrict__ colmax,
                           float* __restrict__ colrcp) {
  __shared__ float xs[NN];
  int b = blockIdx.x;
  int t = threadIdx.x;
  xs[t] = X[b * NN + t];
  xs[t + 256] = X[b * NN + t + 256];
  __syncthreads();
  float c1 = c[0], c2 = c[1], c3 = c[2], c4 = c[3];
  int m = blockIdx.y * 256 + t;
  float xm = xs[m];
  const float* mp = mask + (size_t)b * NN * NN + m;
  float M = -INFINITY, S = 0.f;
  for (int n = 0; n < NN; ++n) {
    float xn = xs[n];
    float val = mp[n * NN] * ((c1 * xn + c3) * xm + (c2 * xn + c4));
    float nM = fmaxf(M, val);
    S = S * __expf(M - nM) + __expf(val - nM);
    M = nM;
  }
  colmax[b * NN + m] = M;
  colrcp[b * NN + m] = 1.0f / S;
}

// K3: s0T[n*128+b] = sum_m soft(n,m); s1T = sum_m soft(n,m)*xm. 16 rows n per block.
__global__ void k_rowsums(const float* __restrict__ X, const float* __restrict__ mask,
                          const float* __restrict__ c, const float* __restrict__ colmax,
                          const float* __restrict__ colrcp, float* __restrict__ s0T,
                          float* __restrict__ s1T) {
  __shared__ float xs[NN], cms[NN], crs[NN];
  __shared__ float red0[8], red1[8];
  int b = blockIdx.x;
  int t = threadIdx.x;
  int lane = t & 31, wv = t >> 5;
  for (int i = t; i < NN; i += 256) {
    xs[i] = X[b * NN + i];
    cms[i] = colmax[b * NN + i];
    crs[i] = colrcp[b * NN + i];
  }
  __syncthreads();
  float c1 = c[0], c2 = c[1], c3 = c[2], c4 = c[3];
  const float* mp = mask + (size_t)b * NN * NN;
  int n0 = blockIdx.y * 16;
  for (int r = 0; r < 16; ++r) {
    int n = n0 + r;
    float xn = xs[n];
    float pn = c1 * xn + c3, qn = c2 * xn + c4;
    float p0 = 0.f, p1 = 0.f;
#pragma unroll
    for (int it = 0; it < 2; ++it) {
      int m = t + it * 256;
      float xm = xs[m];
      float val = mp[n * NN + m] * (pn * xm + qn);
      float e = __expf(val - cms[m]) * crs[m];
      p0 += e;
      p1 += e * xm;
    }
    p0 = wave_sum(p0);
    p1 = wave_sum(p1);
    if (lane == 0) { red0[wv] = p0; red1[wv] = p1; }
    __syncthreads();
    if (t == 0) {
      float t0 = 0.f, t1 = 0.f;
#pragma unroll
      for (int w = 0; w < 8; ++w) { t0 += red0[w]; t1 += red1[w]; }
      s0T[n * BB + b] = t0;
      s1T[n * BB + b] = t1;
    }
    __syncthreads();
  }
}

// K4: Z[b,j] = sum_n s1[b,n]*A[j,n] + s0[b,n]*B[j,n] + bout[j]
// WMMA f32 16x16x4; one 16x16 output tile per wave; K-loop over n (512) x 2 terms.
// Operand layout (wave32): row index = lane&15, k = (lane>>4)*2 + vgpr.
__global__ void k_gemm(const float* __restrict__ s1T, const float* __restrict__ s0T,
                       const float* __restrict__ AT, const float* __restrict__ BT,
                       const float* __restrict__ bout, float* __restrict__ Z) {
  unsigned wave = (blockIdx.x * blockDim.x + threadIdx.x) >> 5;  // 0..255
  int lane = threadIdx.x & 31;
  int tj = wave & 31, tb = wave >> 5;  // 32 j-tiles x 8 b-tiles
  int j0 = tj * 16, b0 = tb * 16;
  int r = lane & 15;
  int hk = (lane >> 4) * 2;
  const float* pS1 = s1T + b0 + r;  // [n][b] layout -> + n*BB
  const float* pS0 = s0T + b0 + r;
  const float* pA = AT + j0 + r;    // [n][j] layout -> + n*NN
  const float* pB = BT + j0 + r;
  v8f acc = {};
  for (int n = 0; n < NN; n += 4) {
    int k0 = n + hk;
    v2f a1; a1.x = pS1[k0 * BB]; a1.y = pS1[(k0 + 1) * BB];
    v2f bA; bA.x = pA[k0 * NN];  bA.y = pA[(k0 + 1) * NN];
    acc = __builtin_amdgcn_wmma_f32_16x16x4_f32(false, a1, false, bA, (short)0, acc,
                                                false, false);
    v2f a0; a0.x = pS0[k0 * BB]; a0.y = pS0[(k0 + 1) * BB];
    v2f bB; bB.x = pB[k0 * NN];  bB.y = pB[(k0 + 1) * NN];
    acc = __builtin_amdgcn_wmma_f32_16x16x4_f32(false, a0, false, bB, (short)0, acc,
                                                false, false);
  }
  float bj = bout[j0 + r];
  int rh = (lane >> 4) * 8;
#pragma unroll
  for (int rr = 0; rr < 8; ++rr) {
    Z[(b0 + rr + rh) * NN + j0 + r] = acc[rr] + bj;
  }
}

extern "C" void kernel_launch(void* const* d_in, const int* in_sizes, int n_in,
                              void* d_out, int out_size, void* d_ws, size_t ws_size,
                              hipStream_t stream) {
  const float* X = (const float*)d_in[0];
  const float* mask = (const float*)d_in[1];
  const float* Wq = (const float*)d_in[2];
  const float* bq = (const float*)d_in[3];
  const float* Wk = (const float*)d_in[4];
  const float* bk = (const float*)d_in[5];
  const float* Wv = (const float*)d_in[6];
  const float* bv = (const float*)d_in[7];
  const float* Wout = (const float*)d_in[8];
  const float* bout = (const float*)d_in[9];
  float* Z = (float*)d_out;

  float* w = (float*)d_ws;
  float* c = w;                      // 4 floats (+ pad to 256)
  float* AT = w + 256;               // 512*512
  float* BT = AT + NN * NN;          // 512*512
  float* colmax = BT + NN * NN;      // 128*512
  float* colrcp = colmax + BB * NN;  // 128*512
  float* s1T = colrcp + BB * NN;     // 512*128 (layout [n][b])
  float* s0T = s1T + BB * NN;        // 512*128

  k_consts<<<1, 32, 0, stream>>>(Wq, bq, Wk, bk, c);
  k_wout<<<(NN * NN) / 8, 256, 0, stream>>>(Wout, Wv, bv, AT, BT);
  k_colstats<<<dim3(BB, 2), 256, 0, stream>>>(X, mask, c, colmax, colrcp);
  k_rowsums<<<dim3(BB, NN / 16), 256, 0, stream>>>(X, mask, c, colmax, colrcp, s0T, s1T);
  k_gemm<<<32, 256, 0, stream>>>(s1T, s0T, AT, BT, bout, Z);
}